// AFM_47802986004583
// MI455X (gfx1250) — compile-verified
//
#include <hip/hip_runtime.h>
#include <math.h>

#define NF 26
#define ND 13
#define NP 325
#define ED 32
#define AD 8
#define VOCAB 100000
#define WAVES 8
#define TPB (WAVES * 32)

typedef __attribute__((ext_vector_type(2))) float v2f;
typedef __attribute__((ext_vector_type(8))) float v8f;

__global__ __launch_bounds__(TPB) void afm_fused_kernel(
    const float* __restrict__ dense,   // [B,13]
    const int*   __restrict__ sparse,  // [B,26]
    const float* __restrict__ tbl,     // [26,100000,32]
    const float* __restrict__ w1,      // [32,8]
    const float* __restrict__ b1,      // [8]
    const float* __restrict__ w2,      // [8,1]
    const float* __restrict__ b2,      // [1]
    const float* __restrict__ fw,      // [45,1]
    const float* __restrict__ fb,      // [1]
    float* __restrict__ out)           // [B]
{
    // k-major embedding tile per wave: emb[k][f], stride 26 -> conflict-free
    __shared__ float embT[WAVES][ED][NF];        // 26624 B
    __shared__ float sc[WAVES][336];             // 10752 B (scores -> softmax weights)
    __shared__ unsigned short pairs[NP];         // (f1<<8)|f2

    const int tid  = threadIdx.x;
    const int wave = tid >> 5;
    const int lane = tid & 31;
    const int b    = blockIdx.x * WAVES + wave;
    const int hi   = lane >> 4;    // K-half select for WMMA fragments
    const int lo   = lane & 15;    // M-row / N-col within fragment

    // ---- pair table (triu_indices of 26, k=1), cooperative init ----
    for (int p = tid; p < NP; p += TPB) {
        int rem = p, f1 = 0;
        while (rem >= (NF - 1 - f1)) { rem -= (NF - 1 - f1); ++f1; }
        pairs[p] = (unsigned short)((f1 << 8) | (f1 + 1 + rem));
    }

    // ---- stage 0: embedding gather -> LDS (coalesced 128B rows) ----
    int idx_l = (lane < NF) ? sparse[b * NF + lane] : 0;
    #pragma unroll
    for (int f = 0; f < NF; ++f) {
        int idx = __shfl(idx_l, f, 32);
        float v = tbl[(size_t)f * ((size_t)VOCAB * ED) + (size_t)idx * ED + lane];
        embT[wave][lane][f] = v;
    }

    // ---- preload B fragments for all 8 K-steps (att_w_w padded N=8->16) ----
    v2f bw[8];
    #pragma unroll
    for (int k0 = 0; k0 < 8; ++k0) {
        int r = 4 * k0 + 2 * hi;
        bw[k0].x = (lo < AD) ? w1[r * AD + lo]       : 0.f;
        bw[k0].y = (lo < AD) ? w1[(r + 1) * AD + lo] : 0.f;
    }
    const float bb = (lo < AD) ? b1[lo] : 0.f;   // hidden bias for column lo
    const float dd = (lo < AD) ? w2[lo] : 0.f;   // second-layer weight
    const float db = b2[0];

    __syncthreads();

    // ---- stage 1: 21 tiles x (8x V_WMMA_F32_16X16X4_F32) -> scores ----
    const float* embW = &embT[wave][0][0];
    for (int t = 0; t < 21; ++t) {
        const int p0 = t * 16;
        const int pm = p0 + lo;                 // this lane's pair row (M)
        const bool valid = (pm < NP);
        unsigned int code = pairs[valid ? pm : 0];
        const int f1 = (int)(code >> 8), f2 = (int)(code & 255);

        v8f c = {};
        #pragma unroll
        for (int k0 = 0; k0 < 8; ++k0) {
            const int kk = 4 * k0 + 2 * hi;     // this lane's K slot
            float e1x = embW[kk * NF + f1];
            float e2x = embW[kk * NF + f2];
            float e1y = embW[(kk + 1) * NF + f1];
            float e2y = embW[(kk + 1) * NF + f2];
            v2f a;
            a.x = valid ? e1x * e2x : 0.f;      // bi on the fly, never spilled
            a.y = valid ? e1y * e2y : 0.f;
            c = __builtin_amdgcn_wmma_f32_16x16x4_f32(
                    false, a, false, bw[k0], (short)0, c, false, false);
        }
        // relu + second layer + reduce across the 16 N-columns (cols>=8 are 0)
        #pragma unroll
        for (int j = 0; j < 8; ++j) {
            float contrib = fmaxf(c[j] + bb, 0.f) * dd;
            contrib += __shfl_xor(contrib, 1, 32);
            contrib += __shfl_xor(contrib, 2, 32);
            contrib += __shfl_xor(contrib, 4, 32);
            contrib += __shfl_xor(contrib, 8, 32);
            int p = p0 + j + hi * 8;            // C row j -> pair p0+j / p0+8+j
            if (lo == 0 && p < NP) sc[wave][p] = contrib + db;
        }
    }

    __syncthreads();

    // ---- stage 2: softmax over 325 pairs (per wave) ----
    float m = -3.402823466e38f;
    #pragma unroll
    for (int tt = 0; tt < 11; ++tt) {
        int p = lane + 32 * tt;
        if (p < NP) m = fmaxf(m, sc[wave][p]);
    }
    #pragma unroll
    for (int msk = 16; msk >= 1; msk >>= 1) m = fmaxf(m, __shfl_xor(m, msk, 32));
    float s = 0.f;
    #pragma unroll
    for (int tt = 0; tt < 11; ++tt) {
        int p = lane + 32 * tt;
        if (p < NP) {
            float e = __expf(sc[wave][p] - m);
            sc[wave][p] = e;                    // store unnormalized weight
            s += e;
        }
    }
    #pragma unroll
    for (int msk = 16; msk >= 1; msk >>= 1) s += __shfl_xor(s, msk, 32);
    const float inv = 1.f / s;                  // fold normalization into stage 3

    __syncthreads();

    // ---- stage 3: att_out[k=lane] = inv * sum_p w_p * e1[k]*e2[k] ----
    float acc = 0.f;
    const float* er = &embT[wave][lane][0];     // lane-k row, stride-26 reads
    for (int p = 0; p < NP; ++p) {
        unsigned int code = pairs[p];           // broadcast
        float w = sc[wave][p];                  // broadcast
        acc = fmaf(w * er[code >> 8], er[code & 255], acc);
    }
    float att = acc * inv;

    // ---- stage 4: final linear + sigmoid ----
    float y = att * fw[lane];
    if (lane < ND) y += dense[b * ND + lane] * fw[ED + lane];
    #pragma unroll
    for (int msk = 16; msk >= 1; msk >>= 1) y += __shfl_xor(y, msk, 32);
    if (lane == 0) {
        float z = y + fb[0];
        out[b] = 1.f / (1.f + __expf(-z));
    }
}

extern "C" void kernel_launch(void* const* d_in, const int* in_sizes, int n_in,
                              void* d_out, int out_size, void* d_ws, size_t ws_size,
                              hipStream_t stream) {
    const float* dense  = (const float*)d_in[0];
    const int*   sparse = (const int*)  d_in[1];
    const float* tbl    = (const float*)d_in[2];
    const float* w1     = (const float*)d_in[3];
    const float* b1     = (const float*)d_in[4];
    const float* w2     = (const float*)d_in[5];
    const float* b2     = (const float*)d_in[6];
    const float* fw     = (const float*)d_in[7];
    const float* fb     = (const float*)d_in[8];
    float* out = (float*)d_out;

    const int B = out_size;                 // 16384
    const int blocks = (B + WAVES - 1) / WAVES;
    afm_fused_kernel<<<blocks, TPB, 0, stream>>>(
        dense, sparse, tbl, w1, b1, w2, b2, fw, fb, out);
}